// Multi_level_Wavelet_Network_Based_on_CNN_Transformer_Hybrid_Attention_21560735826124
// MI455X (gfx1250) — compile-verified
//
#include <hip/hip_runtime.h>
#include <math.h>

typedef __attribute__((ext_vector_type(2))) float v2f;
typedef __attribute__((ext_vector_type(8))) float v8f;

__device__ __forceinline__ float gelu_exact(float v) {
    return 0.5f * v * (1.0f + erff(v * 0.70710678118654752f));
}
__device__ __forceinline__ float sigmoidf_(float v) {
    return 1.0f / (1.0f + expf(-v));
}

// ---------------------------------------------------------------------------
// Kernel 1: per-parity-class sums of Fn (LayerNormed x), deterministic.
// Wave layout: 16 tokens per wave-tile; lane L = (m = L&15 token, h = L>>4
// channel half). Lane holds channels c with c%4 in {2h, 2h+1} (the exact
// V_WMMA_F32_16X16X4_F32 A-matrix striping, reused in kernel 3).
// Each wave owns a fixed (batch, row-parity) slice; col parity == lane&1.
// Per-wave partial sums go to a unique ws slot -> fixed-order reduction later.
// ---------------------------------------------------------------------------
__global__ void k1_parity_sums(const float* __restrict__ x,
                               const int* __restrict__ pH, const int* __restrict__ pW,
                               const float* __restrict__ lnw, const float* __restrict__ lnb,
                               float* __restrict__ partial, int BL)
{
    __shared__ float s_lnw[64], s_lnb[64];
    int tid = threadIdx.x;
    if (tid < 64) { s_lnw[tid] = lnw[tid]; s_lnb[tid] = lnb[tid]; }
    __syncthreads();

    int H = *pH, W = *pW;
    int L = H * W;
    int B = BL / L;
    int totalWaves = BL >> 7;          // each wave covers 128 tokens
    int NW = totalWaves / (2 * B);     // waves per (batch, row-parity)

    int lane = tid & 31, wv = tid >> 5;
    int gid  = blockIdx.x * 8 + wv;
    int b    = gid / (2 * NW);
    int rem  = gid - b * 2 * NW;
    int rp   = rem / NW;               // row parity owned by this wave
    int widx = rem - rp * NW;

    int m = lane & 15, h = lane >> 4;
    int tilesPerRow = W >> 4;
    int tilesPerRP  = L >> 5;          // 16-token tiles per (b, rp)

    v2f acc[16];
#pragma unroll
    for (int k = 0; k < 16; ++k) { acc[k].x = 0.f; acc[k].y = 0.f; }

    const float* xb = x + (size_t)b * L * 64;
    for (int t = widx; t < tilesPerRP; t += NW) {
        int vr  = t / tilesPerRow;
        int j   = t - vr * tilesPerRow;
        int row = 2 * vr + rp;
        int tok = row * W + j * 16 + m;
        const float* p = xb + (size_t)tok * 64 + 2 * h;

        v2f xv[16];
        float s = 0.f, q = 0.f;
#pragma unroll
        for (int k = 0; k < 16; ++k) {
            xv[k] = *reinterpret_cast<const v2f*>(p + 4 * k);
            s += xv[k].x + xv[k].y;
            q += xv[k].x * xv[k].x + xv[k].y * xv[k].y;
        }
        s += __shfl_xor(s, 16, 32);    // combine the two channel-halves
        q += __shfl_xor(q, 16, 32);
        float mu   = s * (1.0f / 64.0f);
        float var  = q * (1.0f / 64.0f) - mu * mu;
        float rstd = rsqrtf(var + 1e-5f);
#pragma unroll
        for (int k = 0; k < 16; ++k) {
            int c0 = 4 * k + 2 * h;
            acc[k].x += (xv[k].x - mu) * rstd * s_lnw[c0]     + s_lnb[c0];
            acc[k].y += (xv[k].y - mu) * rstd * s_lnw[c0 + 1] + s_lnb[c0 + 1];
        }
    }

    // Reduce across lanes sharing (h, col-parity): xor masks 2,4,8 keep
    // parity (bit0) and half (bit4) classes separate. Deterministic order.
#pragma unroll
    for (int k = 0; k < 16; ++k) {
#pragma unroll
        for (int msk = 2; msk <= 8; msk <<= 1) {
            acc[k].x += __shfl_xor(acc[k].x, msk, 32);
            acc[k].y += __shfl_xor(acc[k].y, msk, 32);
        }
    }
    if ((lane & 14) == 0) {            // lanes 0,1,16,17 -> (h, colparity)
        int cp = lane & 1;
        float* dst = partial + (size_t)gid * 128 + cp * 64 + 2 * h;
#pragma unroll
        for (int k = 0; k < 16; ++k)
            *reinterpret_cast<v2f*>(dst + 4 * k) = acc[k];
    }
}

// ---------------------------------------------------------------------------
// Kernel 2: tiny per-batch math. One block per batch (extra blocks exit).
// Fixed-order reduction of wave partials, then M2*M1, w_fuse1, w_fuse2 (gelu),
// w_cs (sigmoid) -> CA[b][64], and gscal[b] = F_global . w_ss[16:32] + b_ss.
// ---------------------------------------------------------------------------
__global__ void k2_globals(const float* __restrict__ partial,
                           const float* __restrict__ w_fuse1, const float* __restrict__ b_fuse1,
                           const float* __restrict__ w_fuse2, const float* __restrict__ b_fuse2,
                           const float* __restrict__ w_cs,    const float* __restrict__ b_cs,
                           const float* __restrict__ w_ss,    const float* __restrict__ b_ss,
                           const int* __restrict__ pH, const int* __restrict__ pW,
                           float* __restrict__ ca_out, float* __restrict__ g_out, int BL)
{
    int H = *pH, W = *pW;
    int L = H * W;
    int B = BL / L;
    int b = blockIdx.x;
    if (b >= B) return;
    int NW = (BL >> 7) / (2 * B);

    __shared__ float pmean[4][64];
    __shared__ float sm[4][64];
    __shared__ float fhp[64];
    __shared__ float Fg[16];
    int tid = threadIdx.x;

    {   // fixed-order partial reduction: pmean[p][c] = mean of Fn over class p
        int p = tid >> 6, c = tid & 63;
        int rp = p >> 1, cp = p & 1;
        const float* base = partial + (size_t)(b * 2 + rp) * NW * 128 + cp * 64 + c;
        float acc = 0.f;
        for (int w = 0; w < NW; ++w) acc += base[(size_t)w * 128];
        pmean[p][c] = acc / (float)(L >> 2);
    }
    __syncthreads();
    {   // smean[k][c] = (M2*M1)[k,:] . pmean[:,c]  (rolls are mean-invariant)
        const float M1[4][4] = {{ 0.9124f,-0.0256f, 0.4083f, 0.0115f},
                                { 0.0256f, 0.9124f, 0.0115f,-0.4083f},
                                {-0.4083f, 0.0115f, 0.9124f, 0.0256f},
                                { 0.0115f, 0.4083f,-0.0256f, 0.9124f}};
        const float M2[4][4] = {{ 1.3066f, 0.5412f, 0.5412f, 1.3066f},
                                {-0.5412f, 1.3066f, 1.3066f,-0.5412f},
                                {-0.4671f,-1.3349f, 1.3349f, 0.4671f},
                                { 1.3349f,-0.4671f, 0.4671f,-1.3349f}};
        int k = tid >> 6, c = tid & 63;
        float m21[4];
        for (int p = 0; p < 4; ++p) {
            float a = 0.f;
            for (int q = 0; q < 4; ++q) a += M2[k][q] * M1[q][p];
            m21[p] = a;
        }
        float v = 0.f;
        for (int p = 0; p < 4; ++p) v += m21[p] * pmean[p][c];
        sm[k][c] = v;
    }
    __syncthreads();
    if (tid < 64) {   // FH123_pool = [mean FH1|FH2|FH3] @ w_fuse1 + b
        float a = b_fuse1[tid];
        for (int j = 0; j < 192; ++j)
            a += sm[1 + (j >> 6)][j & 63] * w_fuse1[j * 64 + tid];
        fhp[tid] = a;
    }
    __syncthreads();
    if (tid < 16) {   // F_global = gelu([FL0_pool|FH123_pool|F_pool] @ w_fuse2 + b)
        float a = b_fuse2[tid];
        for (int j = 0; j < 64; ++j) a += sm[0][j] * w_fuse2[j * 16 + tid];
        for (int j = 0; j < 64; ++j) a += fhp[j]   * w_fuse2[(64 + j) * 16 + tid];
        for (int j = 0; j < 64; ++j) {
            float fp = 0.25f * (pmean[0][j] + pmean[1][j] + pmean[2][j] + pmean[3][j]);
            a += fp * w_fuse2[(128 + j) * 16 + tid];
        }
        Fg[tid] = gelu_exact(a);
    }
    __syncthreads();
    if (tid < 64) {   // F_CA
        float a = b_cs[tid];
        for (int r = 0; r < 16; ++r) a += Fg[r] * w_cs[r * 64 + tid];
        ca_out[b * 64 + tid] = sigmoidf_(a);
    }
    if (tid == 0) {   // global half of the SA logit
        float a = b_ss[0];
        for (int r = 0; r < 16; ++r) a += Fg[r] * w_ss[16 + r];
        g_out[b] = a;
    }
}

// ---------------------------------------------------------------------------
// Kernel 3: per 16-token tile per wave. Recompute LN (x is re-read anyway),
// build WMMA A operands in-register (lane layout == V_WMMA_F32_16X16X4_F32
// A striping), chain 16 fp32 WMMAs for F_local_pre = Fn(16x64) @ w_local(64x16),
// gelu, dot with w_ss via lane butterflies, sigmoid -> SA, then out = x*CA*SA.
// ---------------------------------------------------------------------------
__global__ void k3_output(const float* __restrict__ x,
                          const int* __restrict__ pH, const int* __restrict__ pW,
                          const float* __restrict__ lnw, const float* __restrict__ lnb,
                          const float* __restrict__ w_local, const float* __restrict__ b_local,
                          const float* __restrict__ w_ss,
                          const float* __restrict__ ca_arr, const float* __restrict__ g_arr,
                          float* __restrict__ out, int BL)
{
    __shared__ float s_lnw[64], s_lnb[64], s_ca[64];
    __shared__ float s_sa[8][16];
    int tid = threadIdx.x;
    int H = *pH, W = *pW;
    int L = H * W;
    int tilesPerB = L >> 4;

    int lane  = tid & 31, wv = tid >> 5;
    int tileG = blockIdx.x * 8 + wv;
    int b     = tileG / tilesPerB;
    int tok0  = (tileG - b * tilesPerB) << 4;

    int bb = (blockIdx.x * 8) / tilesPerB;      // whole block lives in one batch
    if (tid < 64) { s_lnw[tid] = lnw[tid]; s_lnb[tid] = lnb[tid]; s_ca[tid] = ca_arr[bb * 64 + tid]; }
    __syncthreads();

    int m = lane & 15, h = lane >> 4;
    const float* xp = x + ((size_t)b * L + tok0 + m) * 64 + 2 * h;

    v2f xv[16];
    float s = 0.f, q = 0.f;
#pragma unroll
    for (int k = 0; k < 16; ++k) {
        xv[k] = *reinterpret_cast<const v2f*>(xp + 4 * k);
        s += xv[k].x + xv[k].y;
        q += xv[k].x * xv[k].x + xv[k].y * xv[k].y;
    }
    s += __shfl_xor(s, 16, 32);
    q += __shfl_xor(q, 16, 32);
    float mu   = s * (1.0f / 64.0f);
    float rstd = rsqrtf(q * (1.0f / 64.0f) - mu * mu + 1e-5f);

    // B operand: w_local (64x16) in the 4x16 KxN WMMA striping per K-step.
    v2f bm[16];
#pragma unroll
    for (int k = 0; k < 16; ++k) {
        int c0 = 4 * k + 2 * h;
        bm[k].x = w_local[c0 * 16 + m];
        bm[k].y = w_local[(c0 + 1) * 16 + m];
    }

    v8f acc;
#pragma unroll
    for (int i = 0; i < 8; ++i) acc[i] = 0.f;

#pragma unroll
    for (int k = 0; k < 16; ++k) {
        int c0 = 4 * k + 2 * h;
        v2f a;
        a.x = (xv[k].x - mu) * rstd * s_lnw[c0]     + s_lnb[c0];
        a.y = (xv[k].y - mu) * rstd * s_lnw[c0 + 1] + s_lnb[c0 + 1];
        // D(16x16 f32) = A(16x4 f32) x B(4x16 f32) + C   [full wave, EXEC all 1s]
        acc = __builtin_amdgcn_wmma_f32_16x16x4_f32(false, a, false, bm[k],
                                                    (short)0, acc, false, false);
    }

    // SA logit: sum_n gelu(pre[t][n] + b_local[n]) * w_ss[n], per token t=v+8h.
    float bl = b_local[m];
    float wn = w_ss[m];
    float g  = g_arr[b];
    float sv[8];
#pragma unroll
    for (int v = 0; v < 8; ++v) {
        float t = gelu_exact(acc[v] + bl) * wn;
        t += __shfl_xor(t, 1, 32);
        t += __shfl_xor(t, 2, 32);
        t += __shfl_xor(t, 4, 32);
        t += __shfl_xor(t, 8, 32);
        sv[v] = sigmoidf_(t + g);
    }
    if (m == 0) {
#pragma unroll
        for (int v = 0; v < 8; ++v) s_sa[wv][h * 8 + v] = sv[v];
    }
    __syncthreads();
    float sa = s_sa[wv][m];

    float* op = out + ((size_t)b * L + tok0 + m) * 64 + 2 * h;
#pragma unroll
    for (int k = 0; k < 16; ++k) {
        int c0 = 4 * k + 2 * h;
        v2f o;
        o.x = xv[k].x * s_ca[c0]     * sa;
        o.y = xv[k].y * s_ca[c0 + 1] * sa;
        *reinterpret_cast<v2f*>(op + 4 * k) = o;
    }
}

// ---------------------------------------------------------------------------
extern "C" void kernel_launch(void* const* d_in, const int* in_sizes, int n_in,
                              void* d_out, int out_size, void* d_ws, size_t ws_size,
                              hipStream_t stream)
{
    const float* x       = (const float*)d_in[0];
    const int*   pH      = (const int*)d_in[1];
    const int*   pW      = (const int*)d_in[2];
    const float* lnw     = (const float*)d_in[3];
    const float* lnb     = (const float*)d_in[4];
    const float* w_fuse1 = (const float*)d_in[5];
    const float* b_fuse1 = (const float*)d_in[6];
    const float* w_fuse2 = (const float*)d_in[7];
    const float* b_fuse2 = (const float*)d_in[8];
    const float* w_local = (const float*)d_in[9];
    const float* b_local = (const float*)d_in[10];
    const float* w_cs    = (const float*)d_in[11];
    const float* b_cs    = (const float*)d_in[12];
    const float* w_ss    = (const float*)d_in[13];
    const float* b_ss    = (const float*)d_in[14];
    float* out = (float*)d_out;
    float* ws  = (float*)d_ws;

    int C  = in_sizes[3];          // channels (64)
    int BL = in_sizes[0] / C;      // total tokens B*L (524288)

    float* partial = ws;           // BL floats (one 128-float slot per wave)
    float* ca      = ws + BL;      // B*C floats
    float* gsa     = ws + BL + 2048;

    int blocks1 = BL / 1024;       // 8 waves/block, 128 tokens/wave
    hipLaunchKernelGGL(k1_parity_sums, dim3(blocks1), dim3(256), 0, stream,
                       x, pH, pW, lnw, lnb, partial, BL);

    hipLaunchKernelGGL(k2_globals, dim3(16), dim3(256), 0, stream,
                       partial, w_fuse1, b_fuse1, w_fuse2, b_fuse2,
                       w_cs, b_cs, w_ss, b_ss, pH, pW, ca, gsa, BL);

    int blocks3 = BL / 128;        // 8 wave-tiles (16 tokens each) per block
    hipLaunchKernelGGL(k3_output, dim3(blocks3), dim3(256), 0, stream,
                       x, pH, pW, lnw, lnb, w_local, b_local, w_ss,
                       ca, gsa, out, BL);
}